// LEsemble_89189290868906
// MI455X (gfx1250) — compile-verified
//
#include <hip/hip_runtime.h>
#include <hip/hip_bf16.h>
#include <math.h>

#define N      1024
#define NB     16
#define NTILE  (N / NB)          // 64
#define BATCH  128
#define TOTMAT (BATCH + 1)       // matrix 0 = L + I, 1..128 = M_b
#define EPSV   1e-8f

typedef __attribute__((ext_vector_type(2))) float v2f;
typedef __attribute__((ext_vector_type(8))) float v8f;

// ---------------------------------------------------------------------------
// Triangular pair index helper: w -> (r, c) with r >= c, w = r*(r+1)/2 + c
// ---------------------------------------------------------------------------
__device__ __forceinline__ int tri_row(int w) {
    int r = (int)((sqrtf(8.0f * (float)w + 1.0f) - 1.0f) * 0.5f);
    while ((r + 1) * (r + 2) / 2 <= w) r++;
    while (r * (r + 1) / 2 > w) r--;
    return r;
}

// ---------------------------------------------------------------------------
// L = B^T B + eps*I   (lower triangle only), one wave per 16x16 tile.
// f32 WMMA 16x16x4: lane<16 holds K=k0,k0+1 ; lane>=16 holds K=k0+2,k0+3.
// ---------------------------------------------------------------------------
__global__ void gemm_btb(const float* __restrict__ B, float* __restrict__ L,
                         int ntiles) {
    const int wave = (int)((blockIdx.x * blockDim.x + threadIdx.x) >> 5);
    const int lane = (int)(threadIdx.x & 31);
    if (wave >= ntiles) return;                 // wave-uniform: EXEC stays full

    const int bi = tri_row(wave);
    const int bk = wave - bi * (bi + 1) / 2;
    const int row0 = bi * NB, col0 = bk * NB;
    const int m  = lane & 15;
    const int hi = lane >> 4;

    v8f acc = {};
    for (int k0 = 0; k0 < N; k0 += 4) {
        const int kk = k0 + 2 * hi;
        v2f a, b;
        a.x = B[(size_t)kk * N + row0 + m];
        a.y = B[(size_t)(kk + 1) * N + row0 + m];
        b.x = B[(size_t)kk * N + col0 + m];
        b.y = B[(size_t)(kk + 1) * N + col0 + m];
        acc = __builtin_amdgcn_wmma_f32_16x16x4_f32(false, a, false, b,
                                                    (short)0, acc, false, false);
    }
#pragma unroll
    for (int v = 0; v < 8; v++) {
        const int r = row0 + v + 8 * hi;
        const int c = col0 + m;
        if (c <= r) {
            float val = acc[v] + ((r == c) ? EPSV : 0.0f);
            L[(size_t)r * N + c] = val;
        }
    }
}

// ---------------------------------------------------------------------------
// Build lower triangles of chunk matrices:
//   global index 0     : L + I
//   global index 1+b   : where(mask_b[i]&mask_b[j], L, I)
// ---------------------------------------------------------------------------
__global__ void build_M(const float* __restrict__ L, const int* __restrict__ x,
                        float* __restrict__ mats, int chunkStart) {
    const int mat = (int)blockIdx.y;
    const long long idx = (long long)blockIdx.x * blockDim.x + threadIdx.x;
    const int i = (int)(idx >> 10);
    const int j = (int)(idx & (N - 1));
    if (i >= N || j > i) return;

    float* A = mats + (size_t)mat * N * N;
    const int g = chunkStart + mat;
    float out;
    if (g == 0) {
        out = L[(size_t)i * N + j] + ((i == j) ? 1.0f : 0.0f);
    } else {
        const int* xb = x + (size_t)(g - 1) * N;
        const bool sel = (xb[i] != 0) & (xb[j] != 0);
        out = sel ? L[(size_t)i * N + j] : ((i == j) ? 1.0f : 0.0f);
    }
    A[(size_t)i * N + j] = out;
}

// ---------------------------------------------------------------------------
// Factor 16x16 diagonal block (j,j) in LDS, one wave per matrix.
// Accumulates 2*sum(log(diag)) into acc[mat].
// ---------------------------------------------------------------------------
__global__ void chol_diag(float* __restrict__ mats, float* __restrict__ acc,
                          int j) {
    const int mat  = (int)blockIdx.x;
    const int lane = (int)threadIdx.x;          // 32 threads = 1 wave
    float* A = mats + (size_t)mat * N * N + (size_t)(j * NB) * N + j * NB;

    __shared__ float blk[NB][NB + 1];
    if (lane < NB) {
        for (int c = 0; c <= lane; c++) blk[lane][c] = A[(size_t)lane * N + c];
    }
    __syncthreads();

    for (int k = 0; k < NB; k++) {
        if (lane == 0) blk[k][k] = sqrtf(blk[k][k]);
        __syncthreads();
        const float d = blk[k][k];
        if (lane > k && lane < NB) blk[lane][k] /= d;
        __syncthreads();
        if (lane > k && lane < NB) {
            const float lik = blk[lane][k];
            for (int c = k + 1; c <= lane; c++) blk[lane][c] -= lik * blk[c][k];
        }
        __syncthreads();
    }

    if (lane < NB) {
        for (int c = 0; c <= lane; c++) A[(size_t)lane * N + c] = blk[lane][c];
    }
    if (lane == 0) {
        float s = 0.0f;
        for (int k = 0; k < NB; k++) s += logf(blk[k][k]);
        acc[mat] += 2.0f * s;                   // single writer per matrix
    }
}

// ---------------------------------------------------------------------------
// Panel TRSM: rows i > panel: A[i, jb] <- A[i, jb] * inv(Ljj^T)
// ---------------------------------------------------------------------------
__global__ void chol_trsm(float* __restrict__ mats, int j, int rows) {
    const int mat = (int)blockIdx.y;
    const int r   = (int)(blockIdx.x * blockDim.x + threadIdx.x);
    if (r >= rows) return;

    float* A = mats + (size_t)mat * N * N;
    const int i = (j + 1) * NB + r;
    const float* Ljj = A + (size_t)(j * NB) * N + j * NB;
    float* ai = A + (size_t)i * N + j * NB;

    float xv[NB];
#pragma unroll
    for (int c = 0; c < NB; c++) {
        float s = ai[c];
        for (int t = 0; t < c; t++) s -= xv[t] * Ljj[(size_t)c * N + t];
        xv[c] = s / Ljj[(size_t)c * N + c];
    }
#pragma unroll
    for (int c = 0; c < NB; c++) ai[c] = xv[c];
}

// ---------------------------------------------------------------------------
// Store one 16x16 tile of -acc into C (lower-only when on the diagonal).
// ---------------------------------------------------------------------------
__device__ __forceinline__ void store_tile(float* __restrict__ A, int base,
                                           int ti, int tk, const v8f& acc,
                                           int m, int hi) {
    float* C = A + (size_t)(base + ti * NB) * N + (base + tk * NB);
    const bool diag = (ti == tk);
#pragma unroll
    for (int v = 0; v < 8; v++) {
        const int rr = v + 8 * hi;
        if (!diag || m <= rr)
            C[(size_t)rr * N + m] -= acc[v];
    }
}

// ---------------------------------------------------------------------------
// Trailing SYRK update via WMMA, 2x2 register-blocked: each wave owns a
// 32x32 block (4 tiles) of the trailing submatrix and computes
//   C_{rc} -= P_r (16xNB) * P_c(16xNB)^T   for the 2x2 tile combinations,
// reusing each A/B operand twice (16 WMMA ops per wave per panel step).
// Edge super-tiles clamp to the last valid 16-row panel (wave-uniform), so
// EXEC is all-ones at every WMMA; the duplicate results are simply not stored.
// ---------------------------------------------------------------------------
__global__ void chol_syrk(float* __restrict__ mats, int j, int nt, int nsp) {
    const int wave = (int)((blockIdx.x * blockDim.x + threadIdx.x) >> 5);
    const int lane = (int)(threadIdx.x & 31);
    if (wave >= nsp) return;                    // wave-uniform

    const int mat = (int)blockIdx.y;
    const int si = tri_row(wave);
    const int sk = wave - si * (si + 1) / 2;

    float* A = mats + (size_t)mat * N * N;
    const int base = (j + 1) * NB;

    const int  ti0 = 2 * si,  tk0 = 2 * sk;
    const bool vR1 = (ti0 + 1 < nt);
    const bool vC1 = (tk0 + 1 < nt);
    const int  ti1 = vR1 ? ti0 + 1 : ti0;       // clamped duplicates
    const int  tk1 = vC1 ? tk0 + 1 : tk0;

    const float* Pi0 = A + (size_t)(base + ti0 * NB) * N + j * NB;
    const float* Pi1 = A + (size_t)(base + ti1 * NB) * N + j * NB;
    const float* Pk0 = A + (size_t)(base + tk0 * NB) * N + j * NB;
    const float* Pk1 = A + (size_t)(base + tk1 * NB) * N + j * NB;

    const int m  = lane & 15;
    const int hi = lane >> 4;
    const size_t off = (size_t)m * N;

    v8f a00 = {}, a01 = {}, a10 = {}, a11 = {};
#pragma unroll
    for (int k0 = 0; k0 < NB; k0 += 4) {
        const int kk = k0 + 2 * hi;
        const v2f ar0 = *(const v2f*)(Pi0 + off + kk);   // contiguous b64
        const v2f ar1 = *(const v2f*)(Pi1 + off + kk);
        const v2f bc0 = *(const v2f*)(Pk0 + off + kk);
        const v2f bc1 = *(const v2f*)(Pk1 + off + kk);
        a00 = __builtin_amdgcn_wmma_f32_16x16x4_f32(false, ar0, false, bc0,
                                                    (short)0, a00, false, false);
        a01 = __builtin_amdgcn_wmma_f32_16x16x4_f32(false, ar0, false, bc1,
                                                    (short)0, a01, false, false);
        a10 = __builtin_amdgcn_wmma_f32_16x16x4_f32(false, ar1, false, bc0,
                                                    (short)0, a10, false, false);
        a11 = __builtin_amdgcn_wmma_f32_16x16x4_f32(false, ar1, false, bc1,
                                                    (short)0, a11, false, false);
    }

    // All store predicates below are wave-uniform except the diagonal mask.
    store_tile(A, base, ti0, tk0, a00, m, hi);                    // always
    if (vC1 && ti0 >= tk0 + 1) store_tile(A, base, ti0, tk0 + 1, a01, m, hi);
    if (vR1)                   store_tile(A, base, ti0 + 1, tk0, a10, m, hi);
    if (vR1 && vC1)            store_tile(A, base, ti0 + 1, tk0 + 1, a11, m, hi);
}

// ---------------------------------------------------------------------------
__global__ void zero_acc(float* acc, int n) {
    const int i = (int)(blockIdx.x * blockDim.x + threadIdx.x);
    if (i < n) acc[i] = 0.0f;
}

__global__ void finalize(const float* __restrict__ acc, float* __restrict__ out,
                         int batch) {
    const int b = (int)(blockIdx.x * blockDim.x + threadIdx.x);
    if (b < batch) out[b] = acc[0] - acc[1 + b];
}

// ---------------------------------------------------------------------------
extern "C" void kernel_launch(void* const* d_in, const int* in_sizes, int n_in,
                              void* d_out, int out_size, void* d_ws, size_t ws_size,
                              hipStream_t stream) {
    const int*   x = (const int*)d_in[0];     // [BATCH, N] int32 0/1
    const float* B = (const float*)d_in[1];   // [N, N] float32
    float* out = (float*)d_out;               // [BATCH] float32

    const size_t matBytes = (size_t)N * N * sizeof(float);
    float* L    = (float*)d_ws;
    float* acc  = (float*)((char*)d_ws + matBytes);            // TOTMAT floats
    float* mats = (float*)((char*)d_ws + matBytes + 1024);

    size_t avail = (ws_size > matBytes + 1024) ? (ws_size - matBytes - 1024) : 0;
    int cap = (int)(avail / matBytes);
    if (cap < 1)  cap = 1;                    // best effort if ws is tiny
    if (cap > 32) cap = 32;                   // keep chunk L2-resident (<=128MB)

    zero_acc<<<1, 256, 0, stream>>>(acc, TOTMAT);

    const int gtiles = NTILE * (NTILE + 1) / 2;                // 2080
    gemm_btb<<<(gtiles + 3) / 4, 128, 0, stream>>>(B, L, gtiles);

    for (int start = 0; start < TOTMAT; start += cap) {
        const int nmat = (TOTMAT - start < cap) ? (TOTMAT - start) : cap;

        dim3 bg((N * N) / 256, nmat);
        build_M<<<bg, 256, 0, stream>>>(L, x, mats, start);

        for (int j = 0; j < NTILE; j++) {
            chol_diag<<<nmat, 32, 0, stream>>>(mats, acc + start, j);

            const int rows = N - (j + 1) * NB;
            if (rows > 0) {
                dim3 tg((rows + 255) / 256, nmat);
                chol_trsm<<<tg, 256, 0, stream>>>(mats, j, rows);

                const int nt  = rows / NB;
                const int nst = (nt + 1) / 2;           // 2x2 super-tiles
                const int nsp = nst * (nst + 1) / 2;    // super-tile pairs
                dim3 sg((nsp + 3) / 4, nmat);
                chol_syrk<<<sg, 128, 0, stream>>>(mats, j, nt, nsp);
            }
        }
    }

    finalize<<<1, 128, 0, stream>>>(acc, out, BATCH);
}